// HierarchicalPoseEncoder_28467043238534
// MI455X (gfx1250) — compile-verified
//
#include <hip/hip_runtime.h>

typedef float v2f __attribute__((ext_vector_type(2)));
typedef float v8f __attribute__((ext_vector_type(8)));

#define NJ  24
#define DF  19
#define DPJ 6

__device__ __constant__ int kParents[NJ] = {
    -1, 0, 0, 0, 1, 2, 3, 4, 5, 6, 7, 8, 9, 9, 9, 12, 13, 14, 16, 17, 18, 19, 20, 21};

__device__ inline void lds_fence() {
  asm volatile("s_wait_dscnt 0" ::: "memory");
}

__device__ inline v8f wmma4(v2f a, v2f b, v8f c) {
  // D(16x16,f32) = A(16x4,f32) * B(4x16,f32) + C
  return __builtin_amdgcn_wmma_f32_16x16x4_f32(
      /*neg_a=*/false, a, /*neg_b=*/false, b,
      /*c_mod=*/(short)0, c, /*reuse_a=*/false, /*reuse_b=*/false);
}

__device__ inline int imin(int a, int b) { return a < b ? a : b; }

__global__ __launch_bounds__(128) void pose_enc_kernel(
    const float* __restrict__ rots, const float* __restrict__ Jtrs,
    const float* __restrict__ W0,   const float* __restrict__ b0,
    const float* __restrict__ W1,   const float* __restrict__ b1,
    const float* __restrict__ W2,   const float* __restrict__ b2,
    float* __restrict__ out, int B)
{
  // Per-wave shuffle tiles (D-fragment <-> A-fragment via LDS)
  __shared__ float s_in[4][16][20];            // in_feat tile, K padded 19->20
  __shared__ float s_h [4][16][20];            // hidden tile, K padded 19->20
  __shared__ float s_feat[4][NJ + 1][16][DPJ]; // joint outputs; [NJ] = global_feat

  // Block-shared weights, pair-packed for B-fragments:
  //   entry [.][p][c] = { W[c][2p], W[c][2p+1] }, zero-padded out of range.
  __shared__ v2f s_w1a[NJ][10][16];  // W1 cols 0..15        (30720 B)
  __shared__ v2f s_w1b[NJ][10][4];   // W1 cols 16..18 (+pad) ( 7680 B)
  __shared__ v2f s_w2p[NJ][10][8];   // W2 cols 0..5   (+pad) (15360 B)
  __shared__ v2f s_w0p[144][8];      // W0 cols 0..5   (+pad) ( 9216 B)

  const int tid  = threadIdx.x;

  // ---- Stage packed weights once per block (branch-free guards) ----
  for (int i = tid; i < NJ * 10 * 16; i += 128) {       // W1 N-tile 0
    int c = i & 15, pr = (i >> 4) % 10, j = i / 160;
    int k = 2 * pr;
    float x = W1[j * 361 + c * 19 + k];                  // k <= 18 always valid
    float y = W1[j * 361 + c * 19 + imin(k + 1, 18)];
    y = (k + 1 < DF) ? y : 0.f;
    v2f v; v.x = x; v.y = y;
    s_w1a[j][pr][c] = v;
  }
  for (int i = tid; i < NJ * 10 * 4; i += 128) {         // W1 N-tile 1 (cols 16..18)
    int c = i & 3, pr = (i >> 2) % 10, j = i / 40;
    int k = 2 * pr, col = imin(16 + c, 18);
    float x = W1[j * 361 + col * 19 + k];
    float y = W1[j * 361 + col * 19 + imin(k + 1, 18)];
    x = (c < 3) ? x : 0.f;
    y = (c < 3 && k + 1 < DF) ? y : 0.f;
    v2f v; v.x = x; v.y = y;
    s_w1b[j][pr][c] = v;
  }
  for (int i = tid; i < NJ * 10 * 8; i += 128) {         // W2 (cols 0..5)
    int c = i & 7, pr = (i >> 3) % 10, j = i / 80;
    int k = 2 * pr, col = imin(c, 5);
    float x = W2[j * 114 + col * 19 + k];
    float y = W2[j * 114 + col * 19 + imin(k + 1, 18)];
    x = (c < DPJ) ? x : 0.f;
    y = (c < DPJ && k + 1 < DF) ? y : 0.f;
    v2f v; v.x = x; v.y = y;
    s_w2p[j][pr][c] = v;
  }
  for (int i = tid; i < 144 * 8; i += 128) {             // W0 (K = 288, cols 0..5)
    int c = i & 7, pr = i >> 3;
    int k = 2 * pr, col = imin(c, 5);
    float x = W0[col * 288 + k];
    float y = W0[col * 288 + k + 1];                      // k+1 <= 287 always
    x = (c < DPJ) ? x : 0.f;
    y = (c < DPJ) ? y : 0.f;
    v2f v; v.x = x; v.y = y;
    s_w0p[pr][c] = v;
  }
  __syncthreads();

  const int w    = tid >> 5;          // wave in block
  const int lane = tid & 31;
  const int half = lane >> 4;         // K-half for A/B fragments
  const int n    = lane & 15;         // N column (and A-row id)
  const int n7   = imin(n, 7);
  const int n3   = imin(n, 3);
  const int rb   = (blockIdx.x * 4 + w) * 16;
  if (rb >= B) return;                // wave-uniform; no barriers after this

  // ---------------- Phase 1: global_feat = [rots|Jtrs] @ W0^T + b0 ----------------
  {
    float bv = b0[imin(n, 5)];
    bv = (n < DPJ) ? bv : 0.f;
    v8f acc;
#pragma unroll
    for (int v = 0; v < 8; ++v) acc[v] = bv;

    const float* rrow = rots + (long)(rb + n) * 216;     // A-row = lane&15
    const float* jrow = Jtrs + (long)(rb + n) * 72;
#pragma unroll 4
    for (int step = 0; step < 54; ++step) {              // K 0..215 (rots)
      int pr = 2 * step + half;
      v2f a = *(const v2f*)(rrow + 2 * pr);
      acc = wmma4(a, s_w0p[pr][n7], acc);
    }
#pragma unroll 4
    for (int step = 54; step < 72; ++step) {             // K 216..287 (Jtrs)
      int pr = 2 * step + half;
      v2f a = *(const v2f*)(jrow + 2 * pr - 216);
      acc = wmma4(a, s_w0p[pr][n7], acc);
    }
    if (n < DPJ) {
#pragma unroll
      for (int v = 0; v < 8; ++v) s_feat[w][NJ][v + 8 * half][n] = acc[v];
    }
  }
  lds_fence();

  // ---------------- Phase 2: 24-joint dependency chain ----------------
  for (int j = 0; j < NJ; ++j) {
    const int p = kParents[j];

    // Build in_feat = [rot(9) | Jtr(3) | bone_l(1) | feat_prev(6) | pad]
    if (half == 0) {               // lanes 0..15: rot for row n
      const float* rp = rots + (long)(rb + n) * 216 + j * 9;
#pragma unroll
      for (int q = 0; q < 9; ++q) s_in[w][n][q] = rp[q];
    } else {                       // lanes 16..31: Jtr, bone length, feat_prev
      const float* jp = Jtrs + (long)(rb + n) * 72;
      float x = jp[j * 3 + 0], y = jp[j * 3 + 1], z = jp[j * 3 + 2];
      int pc = (p >= 0) ? p : 0;                         // clamped, load always safe
      float px = jp[pc * 3 + 0], py = jp[pc * 3 + 1], pz = jp[pc * 3 + 2];
      px = (p >= 0) ? px : 0.f;  py = (p >= 0) ? py : 0.f;  pz = (p >= 0) ? pz : 0.f;
      float dx = x - px, dy = y - py, dz = z - pz;
      float bl = sqrtf(dx * dx + dy * dy + dz * dz);
      s_in[w][n][9]  = x;  s_in[w][n][10] = y;
      s_in[w][n][11] = z;  s_in[w][n][12] = bl;
      const int src = (p >= 0) ? p : NJ;
#pragma unroll
      for (int f = 0; f < DPJ; ++f) s_in[w][n][13 + f] = s_feat[w][src][n][f];
      s_in[w][n][19] = 0.f;        // K pad
    }
    lds_fence();

    // Layer 1: h = relu(in_feat @ W1[j]^T + b1[j]), N=19 -> two N-tiles
    float bA = b1[j * DF + n];                            // n <= 15 < 19: safe
    float bB = b1[j * DF + imin(16 + n, 18)];
    bB = (n < 3) ? bB : 0.f;
    v8f accA, accB;
#pragma unroll
    for (int v = 0; v < 8; ++v) { accA[v] = bA; accB[v] = bB; }
#pragma unroll
    for (int step = 0; step < 5; ++step) {
      int pr = 2 * step + half;
      v2f a = *(const v2f*)&s_in[w][n][2 * pr];
      accA = wmma4(a, s_w1a[j][pr][n],  accA);
      accB = wmma4(a, s_w1b[j][pr][n3], accB);
    }
#pragma unroll
    for (int v = 0; v < 8; ++v) {  // ReLU + D-fragment -> LDS (A layout for layer 2)
      int m = v + 8 * half;
      s_h[w][m][n] = fmaxf(accA[v], 0.f);
      if (n < 3) s_h[w][m][16 + n] = fmaxf(accB[v], 0.f);
      if (n == 3) s_h[w][m][19] = 0.f;   // K pad
    }
    lds_fence();

    // Layer 2: out_j = h @ W2[j]^T + b2[j], N=6 (one padded N-tile)
    float bC = b2[j * DPJ + imin(n, 5)];
    bC = (n < DPJ) ? bC : 0.f;
    v8f acc2;
#pragma unroll
    for (int v = 0; v < 8; ++v) acc2[v] = bC;
#pragma unroll
    for (int step = 0; step < 5; ++step) {
      int pr = 2 * step + half;
      v2f a = *(const v2f*)&s_h[w][n][2 * pr];
      acc2 = wmma4(a, s_w2p[j][pr][n7], acc2);
    }
    if (n < DPJ) {
#pragma unroll
      for (int v = 0; v < 8; ++v) {
        int m = v + 8 * half;
        float val = acc2[v];
        s_feat[w][j][m][n] = val;                                  // for children
        out[(long)(rb + m) * (NJ * DPJ) + j * DPJ + n] = val;      // final output
      }
    }
    lds_fence();
  }
}

extern "C" void kernel_launch(void* const* d_in, const int* in_sizes, int n_in,
                              void* d_out, int out_size, void* d_ws, size_t ws_size,
                              hipStream_t stream) {
  const float* rots = (const float*)d_in[0];
  const float* Jtrs = (const float*)d_in[1];
  const float* W0   = (const float*)d_in[2];
  const float* b0   = (const float*)d_in[3];
  const float* W1   = (const float*)d_in[4];
  const float* b1   = (const float*)d_in[5];
  const float* W2   = (const float*)d_in[6];
  const float* b2   = (const float*)d_in[7];
  float* outp = (float*)d_out;

  int B = in_sizes[0] / (NJ * 9);      // rots is (B, 24, 9)
  int grid = (B + 63) / 64;            // 4 waves/block * 16 rows/wave
  pose_enc_kernel<<<grid, 128, 0, stream>>>(rots, Jtrs, W0, b0, W1, b1, W2, b2,
                                            outp, B);
}